// GaussianRasterizer_90615220011437
// MI455X (gfx1250) — compile-verified
//
#include <hip/hip_runtime.h>
#include <math.h>

#define HH 256
#define WW 256
#define PN 512

typedef float v2f __attribute__((ext_vector_type(2)));
typedef float v8f __attribute__((ext_vector_type(8)));

#if defined(__has_builtin)
# if __has_builtin(__builtin_amdgcn_global_load_async_to_lds_b128) && \
     __has_builtin(__builtin_amdgcn_s_wait_asynccnt)
#  define GS_HAVE_ASYNC_LDS 1
# endif
#endif
#ifndef GS_HAVE_ASYNC_LDS
# define GS_HAVE_ASYNC_LDS 0
#endif

#if GS_HAVE_ASYNC_LDS
typedef int gs_v4i __attribute__((vector_size(16)));              // matches builtin param
typedef __attribute__((address_space(1))) gs_v4i gas_v4i;         // global (prints as __device__)
typedef __attribute__((address_space(3))) gs_v4i las_v4i;         // LDS (prints as __shared__)
__device__ __forceinline__ gas_v4i* gs_to_global(const void* p) {
    return (gas_v4i*)(unsigned long long)p;          // AS1 shares the VA
}
__device__ __forceinline__ las_v4i* gs_to_lds(void* p) {
    // low 32 bits of a generic LDS pointer are the LDS byte offset (ISA 10.2)
    return (las_v4i*)(unsigned)(unsigned long long)p;
}
#endif

__constant__ float kC2[5] = {1.0925484305920792f, -1.0925484305920792f, 0.31539156525252005f,
                             -1.0925484305920792f, 0.5462742152960396f};
__constant__ float kC3[7] = {-0.5900435899266435f, 2.890611442640554f, -0.4570457994644658f,
                             0.3731763325901154f, -0.4570457994644658f, 1.445305721320277f,
                             -0.5900435899266435f};

// ---------------------------------------------------------------------------
// Kernel 1: per-gaussian preprocessing.
// U layout (stride 16 floats): [depth, px, py, conA, conB, conC, opaEff, r, g, b]
// ---------------------------------------------------------------------------
__global__ void gs_preprocess(const float* __restrict__ means,
                              const float* __restrict__ feats,
                              const float* __restrict__ opac,
                              const float* __restrict__ scales,
                              const float* __restrict__ rots,
                              const float* __restrict__ vm,
                              const float* __restrict__ pm,
                              const float* __restrict__ campos,
                              const float* __restrict__ ptanx,
                              const float* __restrict__ ptany,
                              float* __restrict__ U)
{
    int g = blockIdx.x * blockDim.x + threadIdx.x;
    if (g >= PN) return;

    const float tanx = ptanx[0], tany = ptany[0];
    const float mx = means[g * 3 + 0], my = means[g * 3 + 1], mz = means[g * 3 + 2];

    // t = Rv @ mean + tv
    const float t0 = vm[0] * mx + vm[1] * my + vm[2] * mz + vm[3];
    const float t1 = vm[4] * mx + vm[5] * my + vm[6] * mz + vm[7];
    const float t2 = vm[8] * mx + vm[9] * my + vm[10] * mz + vm[11];
    const float depth = t2;

    // quaternion -> rotation
    float qw = rots[g * 4 + 0], qx = rots[g * 4 + 1], qy = rots[g * 4 + 2], qz = rots[g * 4 + 3];
    const float qn = rsqrtf(qw * qw + qx * qx + qy * qy + qz * qz);
    qw *= qn; qx *= qn; qy *= qn; qz *= qn;
    float R[3][3] = {
        {1.f - 2.f * (qy * qy + qz * qz), 2.f * (qx * qy - qw * qz), 2.f * (qx * qz + qw * qy)},
        {2.f * (qx * qy + qw * qz), 1.f - 2.f * (qx * qx + qz * qz), 2.f * (qy * qz - qw * qx)},
        {2.f * (qx * qz - qw * qy), 2.f * (qy * qz + qw * qx), 1.f - 2.f * (qx * qx + qy * qy)}};

    const float s[3] = {scales[g * 3 + 0], scales[g * 3 + 1], scales[g * 3 + 2]};
    float M[3][3];
    #pragma unroll
    for (int i = 0; i < 3; ++i)
        #pragma unroll
        for (int j = 0; j < 3; ++j)
            M[i][j] = R[i][j] * s[j];

    float cov3[3][3];
    #pragma unroll
    for (int i = 0; i < 3; ++i)
        #pragma unroll
        for (int j = 0; j < 3; ++j)
            cov3[i][j] = M[i][0] * M[j][0] + M[i][1] * M[j][1] + M[i][2] * M[j][2];

    // Jacobian of perspective projection
    const float fx = WW / (2.f * tanx), fy = HH / (2.f * tany);
    const float limx = 1.3f * tanx, limy = 1.3f * tany;
    const float invz = 1.f / depth;
    const float txz = fminf(fmaxf(t0 * invz, -limx), limx) * depth;
    const float tyz = fminf(fmaxf(t1 * invz, -limy), limy) * depth;
    const float J00 = fx * invz, J02 = -fx * txz * invz * invz;
    const float J11 = fy * invz, J12 = -fy * tyz * invz * invz;

    // T2 = J @ Rv
    float T0[3], T1[3];
    #pragma unroll
    for (int j = 0; j < 3; ++j) {
        T0[j] = J00 * vm[0 * 4 + j] + J02 * vm[2 * 4 + j];
        T1[j] = J11 * vm[1 * 4 + j] + J12 * vm[2 * 4 + j];
    }
    float tmp0[3], tmp1[3];
    #pragma unroll
    for (int l = 0; l < 3; ++l) {
        tmp0[l] = T0[0] * cov3[0][l] + T0[1] * cov3[1][l] + T0[2] * cov3[2][l];
        tmp1[l] = T1[0] * cov3[0][l] + T1[1] * cov3[1][l] + T1[2] * cov3[2][l];
    }
    float a = tmp0[0] * T0[0] + tmp0[1] * T0[1] + tmp0[2] * T0[2] + 0.3f;
    float b = tmp0[0] * T1[0] + tmp0[1] * T1[1] + tmp0[2] * T1[2];
    float c = tmp1[0] * T1[0] + tmp1[1] * T1[1] + tmp1[2] * T1[2] + 0.3f;

    const float det = a * c - b * b;
    const float invdet = 1.f / ((det > 0.f) ? det : 1.f);
    const float conA = c * invdet;
    const float conB = -b * invdet;
    const float conC = a * invdet;

    // screen-space center
    const float pp0 = pm[0] * mx + pm[1] * my + pm[2] * mz + pm[3];
    const float pp1 = pm[4] * mx + pm[5] * my + pm[6] * mz + pm[7];
    const float pp3 = pm[12] * mx + pm[13] * my + pm[14] * mz + pm[15];
    const float pw = pp3 + 1e-7f;
    const float px = ((pp0 / pw + 1.f) * WW - 1.f) * 0.5f;
    const float py = ((pp1 / pw + 1.f) * HH - 1.f) * 0.5f;

    // SH color
    float dx = mx - campos[0], dy = my - campos[1], dz = mz - campos[2];
    const float dn = rsqrtf(dx * dx + dy * dy + dz * dz);
    const float x = dx * dn, y = dy * dn, z = dz * dn;
    const float xx = x * x, yy = y * y, zz = z * z;
    const float xy = x * y, yz = y * z, xz = x * z;
    float bas[16];
    bas[0] = 0.28209479177387814f;
    bas[1] = -0.4886025119029199f * y;
    bas[2] = 0.4886025119029199f * z;
    bas[3] = -0.4886025119029199f * x;
    bas[4] = kC2[0] * xy;
    bas[5] = kC2[1] * yz;
    bas[6] = kC2[2] * (2.f * zz - xx - yy);
    bas[7] = kC2[3] * xz;
    bas[8] = kC2[4] * (xx - yy);
    bas[9] = kC3[0] * y * (3.f * xx - yy);
    bas[10] = kC3[1] * xy * z;
    bas[11] = kC3[2] * y * (4.f * zz - xx - yy);
    bas[12] = kC3[3] * z * (2.f * zz - 3.f * xx - 3.f * yy);
    bas[13] = kC3[4] * x * (4.f * zz - xx - yy);
    bas[14] = kC3[5] * z * (xx - yy);
    bas[15] = kC3[6] * x * (xx - 3.f * yy);

    float col[3];
    #pragma unroll
    for (int ch = 0; ch < 3; ++ch) {
        float acc = 0.f;
        #pragma unroll
        for (int k = 0; k < 16; ++k) acc += bas[k] * feats[g * 48 + k * 3 + ch];
        col[ch] = fmaxf(acc + 0.5f, 0.f);
    }

    const float valid = (depth > 0.2f && det > 0.f) ? 1.f : 0.f;
    float* u = &U[g * 16];
    u[0] = depth; u[1] = px; u[2] = py;
    u[3] = conA;  u[4] = conB; u[5] = conC;
    u[6] = valid * opac[g];
    u[7] = col[0]; u[8] = col[1]; u[9] = col[2];
}

// ---------------------------------------------------------------------------
// Kernel 2: stable O(P^2) rank sort by depth + scatter into S.
// S layout (stride 12 floats): [px, py, conA, conB, conC, opaEff, r, g, b]
// ---------------------------------------------------------------------------
__global__ void gs_sort(const float* __restrict__ U, float* __restrict__ S)
{
    __shared__ float sd[PN];
    const int tid = threadIdx.x;
    sd[tid]       = U[tid * 16];
    sd[tid + 256] = U[(tid + 256) * 16];
    __syncthreads();

    const int i = blockIdx.x * 256 + tid;
    const float di = sd[i];
    int rank = 0;
    for (int j = 0; j < PN; ++j) {
        const float dj = sd[j];
        rank += (dj < di) || (dj == di && j < i);   // stable (argsort) tie-break
    }
    const float* u = &U[i * 16];
    float* so = &S[rank * 12];
    #pragma unroll
    for (int k = 0; k < 9; ++k) so[k] = u[k + 1];
}

// ---------------------------------------------------------------------------
// Kernel 3: tiled alpha blending. One workgroup = 16x8 pixel tile, 8 waves.
// Sorted table staged into LDS via GLOBAL_LOAD_ASYNC_TO_LDS_B128 (ASYNCcnt),
// then transformed into tile-centered coefficients. Each wave: N=16 pixels,
// gaussians in batches of M=16, power via V_WMMA_F32_16X16X4_F32 (K=6 pad 2x4).
// LDS entry (stride 12): [k0..k5, pad, pad, opa, r, g, b].
// ---------------------------------------------------------------------------
__global__ void __launch_bounds__(256) gs_render(const float* __restrict__ S,
                                                 float* __restrict__ out)
{
    __shared__ float sm[PN * 12];
    const int tid = threadIdx.x;
    const int tileX = blockIdx.x & 15;
    const int tileY = blockIdx.x >> 4;
    const float x0 = (float)(tileX * 16);
    const float y0 = (float)(tileY * 8);

#if GS_HAVE_ASYNC_LDS
    __shared__ float stage[PN * 12];                  // raw copy of S (24 KB)
    {
        char* lbase = (char*)&stage[0];
        const char* gbase = (const char*)S;
        #pragma unroll
        for (int k = 0; k < 6; ++k) {                 // 1536 x 16B chunks / 256 thr
            const int chunk = tid + 256 * k;
            __builtin_amdgcn_global_load_async_to_lds_b128(
                gs_to_global(gbase + chunk * 16),
                gs_to_lds(lbase + chunk * 16),
                /*offset=*/0, /*cpol=*/0);
        }
        __builtin_amdgcn_s_wait_asynccnt(0);
    }
    __syncthreads();
    const float* __restrict__ Ssrc = stage;
#else
    const float* __restrict__ Ssrc = S;
#endif

    for (int e = tid; e < PN; e += 256) {
        const float* s = &Ssrc[e * 12];
        const float px = s[0] - x0, py = s[1] - y0;       // tile-centered
        const float A = s[2], B = s[3], C = s[4];
        float* d = &sm[e * 12];
        d[0] = -0.5f * A;
        d[1] = -0.5f * C;
        d[2] = -B;
        d[3] = A * px + B * py;
        d[4] = C * py + B * px;
        d[5] = -0.5f * (A * px * px + C * py * py) - B * px * py;
        d[6] = 0.f; d[7] = 0.f;
        d[8] = s[5]; d[9] = s[6]; d[10] = s[7]; d[11] = s[8];
    }
    __syncthreads();

    const int w    = tid >> 5;       // wave id -> pixel row inside tile
    const int lane = tid & 31;
    const int n    = lane & 15;      // pixel column inside tile (and B/D column)
    const int half = lane >> 4;      // K/M half selector per WMMA layout
    const float lx = (float)n, ly = (float)w;

    // B matrix (4x16, pixel features), constant over the whole loop.
    // lanes 0-15 hold K={0,1}, lanes 16-31 hold K={2,3} of each chunk.
    v2f b0, b1;
    b0.x = half ? (lx * ly) : (lx * lx);
    b0.y = half ? lx        : (ly * ly);
    b1.x = half ? 0.f : ly;
    b1.y = half ? 0.f : 1.f;

    float T = 1.f, aR = 0.f, aG = 0.f, aB = 0.f;

    for (int base = 0; base < PN; base += 16) {
        // A matrix (16 gaussians x 4): row = gaussian (lane&15).
        const float* ea = &sm[(base + n) * 12];
        v2f a0, a1;
        a0.x = ea[2 * half];                  // chunk0: k0,k1 | k2,k3
        a0.y = ea[2 * half + 1];
        a1.x = half ? 0.f : ea[4];            // chunk1: k4,k5 | 0,0
        a1.y = half ? 0.f : ea[5];

        v8f dm = {};
        dm = __builtin_amdgcn_wmma_f32_16x16x4_f32(false, a1, false, b1, (short)0, dm, false, false);
        dm = __builtin_amdgcn_wmma_f32_16x16x4_f32(false, a0, false, b0, (short)0, dm, false, false);
        // dm[r] = power of gaussian (base + half*8 + r) at pixel (x0+n, y0+w)

        float alpha[8];
        float4 cq[8];
        #pragma unroll
        for (int r = 0; r < 8; ++r) {
            const int gg = base + half * 8 + r;
            const float4 q = *(const float4*)&sm[gg * 12 + 8];   // [opa, r, g, b]
            cq[r] = q;
            const float p  = dm[r];
            const float ar = q.x * __expf(p);
            const float al = fminf(0.99f, ar);
            alpha[r] = (p <= 0.f && ar >= (1.f / 255.f)) ? al : 0.f;
        }

        // in-register exclusive prefix product of (1 - alpha)
        float pre[8], run = 1.f;
        #pragma unroll
        for (int r = 0; r < 8; ++r) { pre[r] = run; run *= (1.f - alpha[r]); }

        // stitch lower/upper gaussian halves across the lane pair (n, n+16)
        const float totLo = __shfl(run, n, 32);
        const float totHi = __shfl(run, n + 16, 32);
        const float baseT = T * (half ? totLo : 1.f);

        #pragma unroll
        for (int r = 0; r < 8; ++r) {
            const float wr = alpha[r] * (baseT * pre[r]);
            aR += wr * cq[r].y;
            aG += wr * cq[r].z;
            aB += wr * cq[r].w;
        }
        T *= totLo * totHi;
    }

    // combine partial color sums held by the (n, n+16) lane pair
    const float fR = __shfl(aR, n, 32) + __shfl(aR, n + 16, 32);
    const float fG = __shfl(aG, n, 32) + __shfl(aG, n + 16, 32);
    const float fB = __shfl(aB, n, 32) + __shfl(aB, n + 16, 32);

    if (half == 0) {
        const int x = tileX * 16 + n;
        const int y = tileY * 8 + w;
        const int pix = y * WW + x;
        out[0 * HH * WW + pix] = fR + T;   // bg = 1
        out[1 * HH * WW + pix] = fG + T;
        out[2 * HH * WW + pix] = fB + T;
    }
}

// ---------------------------------------------------------------------------
extern "C" void kernel_launch(void* const* d_in, const int* in_sizes, int n_in,
                              void* d_out, int out_size, void* d_ws, size_t ws_size,
                              hipStream_t stream)
{
    (void)in_sizes; (void)n_in; (void)out_size; (void)ws_size;
    const float* means  = (const float*)d_in[0];
    const float* feats  = (const float*)d_in[1];
    const float* opac   = (const float*)d_in[2];
    const float* scales = (const float*)d_in[3];
    const float* rots   = (const float*)d_in[4];
    const float* vm     = (const float*)d_in[5];
    const float* pm     = (const float*)d_in[6];
    const float* campos = (const float*)d_in[7];
    const float* ptanx  = (const float*)d_in[8];
    const float* ptany  = (const float*)d_in[9];
    float* out = (float*)d_out;

    float* U = (float*)d_ws;          // PN*16 floats
    float* S = U + PN * 16;           // PN*12 floats

    gs_preprocess<<<(PN + 255) / 256, 256, 0, stream>>>(
        means, feats, opac, scales, rots, vm, pm, campos, ptanx, ptany, U);
    gs_sort<<<PN / 256, 256, 0, stream>>>(U, S);
    gs_render<<<(HH / 8) * (WW / 16), 256, 0, stream>>>(S, out);
}